// HierDDmx_28355374088369
// MI455X (gfx1250) — compile-verified
//
#include <hip/hip_runtime.h>
#include <math.h>

typedef __attribute__((ext_vector_type(2))) float v2f;
typedef __attribute__((ext_vector_type(8))) float v8f;

#define LN_EPS 1e-5f

// ---------------------------------------------------------------------------
// GEMM: C[M,N] = A[M,K] @ W[N,K]^T  via V_WMMA_F32_16X16X4_F32.
// Each wave32 owns an (MT*16) x (NT*16) C tile = MT*NT independent f32
// accumulators, so the WMMA group per k-step is hazard-free back-to-back.
// Fragments for step k+4 are preloaded before the WMMAs of step k issue
// (software pipeline), and all addresses are uniform-base + 32-bit lane
// offsets so loads use the saddr+voffset form instead of 64-bit VALU chains.
// A-fragment layout (ISA 16x4 f32): lane&15 = row, (lane>>4)*2 = K-pair.
// ---------------------------------------------------------------------------
template <int MT, int NT>
__global__ __launch_bounds__(128) void wmma_gemm_f32_kernel(
    const float* __restrict__ A,     // M x K
    const float* __restrict__ W,     // N x K
    float* __restrict__ C,           // M x N
    int M, int N, int K) {
    const int wave = threadIdx.x >> 5;
    const int lane = threadIdx.x & 31;
    const int mBase = (blockIdx.y * 4 + wave) * (16 * MT);
    const int nBase = blockIdx.x * (16 * NT);
    if (mBase + 16 * MT > M) return;

    const int l15 = lane & 15;
    const int kp  = (lane >> 4) * 2;

    int aoff[MT];
#pragma unroll
    for (int u = 0; u < MT; ++u) aoff[u] = (mBase + u * 16 + l15) * K + kp;
    int boff[NT];
#pragma unroll
    for (int t = 0; t < NT; ++t) boff[t] = (nBase + t * 16 + l15) * K + kp;

    v8f acc[MT][NT] = {};
    v2f a[MT], b[NT], an[MT], bn[NT];

#pragma unroll
    for (int u = 0; u < MT; ++u) a[u] = *(const v2f*)(A + aoff[u]);
#pragma unroll
    for (int t = 0; t < NT; ++t) b[t] = *(const v2f*)(W + boff[t]);

    for (int k0 = 4; k0 < K; k0 += 4) {
        // issue next-step loads first: their wait lands after 8 WMMAs
#pragma unroll
        for (int u = 0; u < MT; ++u) an[u] = *(const v2f*)(A + aoff[u] + k0);
#pragma unroll
        for (int t = 0; t < NT; ++t) bn[t] = *(const v2f*)(W + boff[t] + k0);
#pragma unroll
        for (int u = 0; u < MT; ++u)
#pragma unroll
            for (int t = 0; t < NT; ++t)
                acc[u][t] = __builtin_amdgcn_wmma_f32_16x16x4_f32(
                    false, a[u], false, b[t], (short)0, acc[u][t], false, false);
#pragma unroll
        for (int u = 0; u < MT; ++u) a[u] = an[u];
#pragma unroll
        for (int t = 0; t < NT; ++t) b[t] = bn[t];
    }
    // peeled final k-step
#pragma unroll
    for (int u = 0; u < MT; ++u)
#pragma unroll
        for (int t = 0; t < NT; ++t)
            acc[u][t] = __builtin_amdgcn_wmma_f32_16x16x4_f32(
                false, a[u], false, b[t], (short)0, acc[u][t], false, false);

    // C/D layout: vgpr r -> row = base + r + 8*(lane>>4), col = nBase+t*16+(lane&15)
    const int rowOff = (lane >> 4) * 8;
#pragma unroll
    for (int u = 0; u < MT; ++u)
#pragma unroll
        for (int t = 0; t < NT; ++t) {
            const int co = (mBase + u * 16 + rowOff) * N + nBase + t * 16 + l15;
#pragma unroll
            for (int r = 0; r < 8; ++r)
                C[co + r * N] = acc[u][t][r];
        }
}

// ---------------------------------------------------------------------------
// RN epilogue: one wave32 per row of 512.
//   n    = LN(H[row,:]) * g + b
//   scal = dot(n, Bmat[row%BAS, :])
//   out  = scal * Amat[:, row%BAS] + Res[row,:]
// ---------------------------------------------------------------------------
__global__ void rn_post_kernel(const float* __restrict__ H,
                               const float* __restrict__ Res,
                               const float* __restrict__ g,
                               const float* __restrict__ bta,
                               const float* __restrict__ Amat,   // (D, BAS)
                               const float* __restrict__ Bmat,   // (BAS, D)
                               float* __restrict__ Out,
                               int M, int D, int BAS) {
    const int row = blockIdx.x * blockDim.y + threadIdx.y;
    if (row >= M) return;
    const int lane = threadIdx.x;           // 0..31
    const int per  = D >> 5;                // 16 for D=512

    const float* h = H + (size_t)row * D;
    float loc[16];
    float s = 0.f, s2 = 0.f;
    for (int i = 0; i < per; ++i) {
        float v = h[lane + i * 32];
        loc[i] = v; s += v; s2 += v * v;
    }
    for (int off = 16; off; off >>= 1) {
        s  += __shfl_xor(s,  off, 32);
        s2 += __shfl_xor(s2, off, 32);
    }
    const float invD = 1.f / (float)D;
    const float mean = s * invD;
    const float var  = s2 * invD - mean * mean;
    const float rs   = rsqrtf(var + LN_EPS);

    const int j = row & (BAS - 1);
    const float* brow = Bmat + (size_t)j * D;
    float dot = 0.f;
    for (int i = 0; i < per; ++i) {
        const int o = lane + i * 32;
        float n = (loc[i] - mean) * rs * g[o] + bta[o];
        dot += n * brow[o];
    }
    for (int off = 16; off; off >>= 1) dot += __shfl_xor(dot, off, 32);

    const float* res = Res + (size_t)row * D;
    float* out = Out + (size_t)row * D;
    for (int i = 0; i < per; ++i) {
        const int o = lane + i * 32;
        out[o] = dot * Amat[(size_t)o * BAS + j] + res[o];
    }
}

// ---------------------------------------------------------------------------
// TS epilogue: one 64-thread block per sequence position s.
//   t[b,:] = LN(T[b*S+s,:]) * g + b          (64-wide LN, both batch rows)
//   Nk[b,i] = sum_j (sum_g P[i,j,g]*cos(2*pi*s/(i*512+j*8+g+2))) * t[b,j]
//   H2 = Nk + Res2
// Wp is generated on the fly (no 33MB materialization); thread = channel i.
// ---------------------------------------------------------------------------
__global__ void ts_post_kernel(const float* __restrict__ T,
                               const float* __restrict__ Res2,
                               const float* __restrict__ g,
                               const float* __restrict__ bta,
                               const float* __restrict__ P,     // (64,64,8)
                               float* __restrict__ H2,
                               int S) {
    const int s   = blockIdx.x;
    const int tid = threadIdx.x;            // 0..63
    __shared__ float t0[64], t1[64];
    __shared__ float stats[4];              // mean0, rs0, mean1, rs1

    const int m0 = s, m1 = S + s;           // batch rows (B==2)
    const float r0 = T[m0 * 64 + tid];
    const float r1 = T[m1 * 64 + tid];
    t0[tid] = r0; t1[tid] = r1;
    __syncthreads();

    const int wave = tid >> 5, lane = tid & 31;
    {   // wave 0 reduces row0, wave 1 reduces row1
        const float* rw = (wave == 0) ? t0 : t1;
        float a = rw[lane], b = rw[lane + 32];
        float ssum = a + b, ssq = a * a + b * b;
        for (int off = 16; off; off >>= 1) {
            ssum += __shfl_xor(ssum, off, 32);
            ssq  += __shfl_xor(ssq,  off, 32);
        }
        if (lane == 0) {
            float mean = ssum * (1.f / 64.f);
            float var  = ssq * (1.f / 64.f) - mean * mean;
            stats[wave * 2]     = mean;
            stats[wave * 2 + 1] = rsqrtf(var + LN_EPS);
        }
    }
    __syncthreads();
    const float mu0 = stats[0], rs0 = stats[1];
    const float mu1 = stats[2], rs1 = stats[3];
    const float gg = g[tid], bb = bta[tid];
    __syncthreads();
    t0[tid] = (r0 - mu0) * rs0 * gg + bb;
    t1[tid] = (r1 - mu1) * rs1 * gg + bb;
    __syncthreads();

    const int i = tid;
    const float tps = 6.283185307179586f * (float)s;
    float nk0 = 0.f, nk1 = 0.f;
    for (int j = 0; j < 64; ++j) {
        const float* prow = P + ((size_t)i * 64 + j) * 8;
        const float pbase = (float)(i * 512 + j * 8 + 2);
        float w = 0.f;
#pragma unroll
        for (int gi = 0; gi < 8; ++gi)
            w += prow[gi] * cosf(tps / (pbase + (float)gi));
        nk0 += w * t0[j];
        nk1 += w * t1[j];
    }
    H2[m0 * 64 + i] = nk0 + Res2[m0 * 64 + i];
    H2[m1 * 64 + i] = nk1 + Res2[m1 * 64 + i];
}

// ---------------------------------------------------------------------------
// Orchestration
// ---------------------------------------------------------------------------
extern "C" void kernel_launch(void* const* d_in, const int* in_sizes, int n_in,
                              void* d_out, int out_size, void* d_ws, size_t ws_size,
                              hipStream_t stream) {
    (void)in_sizes; (void)n_in; (void)out_size; (void)ws_size;

    const float* x  = (const float*)d_in[0];
    const float* W1 = (const float*)d_in[1];
    const float* g1 = (const float*)d_in[2];
    const float* b1 = (const float*)d_in[3];
    const float* A1 = (const float*)d_in[4];
    const float* B1 = (const float*)d_in[5];
    const float* M2 = (const float*)d_in[6];
    const float* g2 = (const float*)d_in[7];
    const float* b2 = (const float*)d_in[8];
    const float* P2 = (const float*)d_in[9];
    const float* R2 = (const float*)d_in[10];
    const float* W3 = (const float*)d_in[11];
    const float* g3 = (const float*)d_in[12];
    const float* b3 = (const float*)d_in[13];
    const float* A3 = (const float*)d_in[14];
    const float* B3 = (const float*)d_in[15];
    const float* R3 = (const float*)d_in[16];
    float* out = (float*)d_out;

    const int S = 2048, M = 4096;           // B=2
    const int d0 = 512, d1 = 64, d2 = 512;

    // Workspace layout (floats); bufA reused H1->U, bufB reused h1->res3.
    float* ws   = (float*)d_ws;
    float* bufA = ws;                        // M*512  (H1, later U)
    float* bufB = ws + (size_t)M * 512;      // M*512  (h1, later res3)
    float* Tbuf = bufB + (size_t)M * 512;    // M*64
    float* res2 = Tbuf + (size_t)M * 64;     // M*64
    float* h2   = res2 + (size_t)M * 64;     // M*64

    const dim3 blkG(128);                    // 4 waves / block
    const dim3 blkRN(32, 4);

    // Layer 1: H1 = x @ W1^T  (4096x512x512), then RN epilogue (residual = x)
    wmma_gemm_f32_kernel<2, 4><<<dim3(d0 / 64, M / 128), blkG, 0, stream>>>(x, W1, bufA, M, d0, d0);
    rn_post_kernel<<<dim3(M / 4), blkRN, 0, stream>>>(bufA, x, g1, b1, A1, B1, bufB, M, d0, 64);

    // Layer 2: T = h1 @ M2^T, res2 = h1 @ R2^T  (4096x64x512)
    wmma_gemm_f32_kernel<2, 4><<<dim3(1, M / 128), blkG, 0, stream>>>(bufB, M2, Tbuf, M, d1, d0);
    wmma_gemm_f32_kernel<2, 4><<<dim3(1, M / 128), blkG, 0, stream>>>(bufB, R2, res2, M, d1, d0);
    ts_post_kernel<<<dim3(S), dim3(64), 0, stream>>>(Tbuf, res2, g2, b2, P2, h2, S);

    // Layer 3: U = h2 @ W3^T, res3 = h2 @ R3^T  (4096x512x64), RN epilogue
    wmma_gemm_f32_kernel<2, 4><<<dim3(d2 / 64, M / 128), blkG, 0, stream>>>(h2, W3, bufA, M, d2, d1);
    wmma_gemm_f32_kernel<2, 4><<<dim3(d2 / 64, M / 128), blkG, 0, stream>>>(h2, R3, bufB, M, d2, d1);
    rn_post_kernel<<<dim3(M / 4), blkRN, 0, stream>>>(bufA, bufB, g3, b3, A3, B3, out, M, d2, 64);
}